// LSAM_71330816852361
// MI455X (gfx1250) — compile-verified
//
#include <hip/hip_runtime.h>
#include <math.h>

typedef __attribute__((ext_vector_type(2))) float v2f;
typedef __attribute__((ext_vector_type(8))) float v8f;

#define TILE_STRIDE 514   // 512 K-features + 2 pad floats -> conflict-free LDS

// ---------------------------------------------------------------------------
// Kernel 1: per (b,d2,h2) stage the 32 (w2) x 512 (c,kd,kh,kw) block matrix
// into LDS, then WMMA f32 16x16x4 against [512 x 16] weights (cols 0=mlp,
// 1=conv, rest zero). K split across 4 wave-pairs, reduced via ds_add_f32.
// ---------------------------------------------------------------------------
__global__ __launch_bounds__(256) void patch_gemm_wmma(
    const float* __restrict__ x,
    const float* __restrict__ mlp_w, const float* __restrict__ mlp_b,
    const float* __restrict__ conv_w, const float* __restrict__ conv_b,
    float* __restrict__ y)
{
  __shared__ float tile[32 * TILE_STRIDE]; // [w2][k]
  __shared__ float wlds[3 * 512];          // row 0: mlp_w, row 1: conv_w, row 2: zeros
  __shared__ float acc[64];                // [n][w2], initialized with biases

  const int t  = threadIdx.x;
  const int wg = blockIdx.x;               // b*512 + d2*32 + h2
  const int h2 = wg & 31;
  const int d2 = (wg >> 5) & 15;
  const int b  = wg >> 9;

  // stage weights (feature order c*8+kd*4+kh*2+kw == flat conv_w order)
  for (int i = t; i < 512; i += 256) {
    wlds[i]        = mlp_w[i];
    wlds[512 + i]  = conv_w[i];
    wlds[1024 + i] = 0.0f;
  }
  if (t < 64) acc[t] = (t < 32) ? mlp_b[0] : conv_b[0];

  // stage x: 256 rows (c,kd,kh), 64 contiguous floats each; lane w2 takes a b64
  const int w2    = t & 31;
  const int rbase = t >> 5;                // wave id -> uniform row group
  const float* xb = x + (size_t)b * (64u * 32u * 64u * 64u);
  #pragma unroll 4
  for (int i = 0; i < 32; ++i) {
    int r  = rbase * 32 + i;
    int c  = r >> 2;
    int kd = (r >> 1) & 1;
    int kh = r & 1;
    const float* gp = xb + (size_t)c * (32 * 64 * 64)
                         + (size_t)(2 * d2 + kd) * (64 * 64)
                         + (2 * h2 + kh) * 64 + 2 * w2;
    v2f v = *(const v2f*)gp;                       // coalesced global b64
    int kbase = c * 8 + kd * 4 + kh * 2;
    *(v2f*)&tile[w2 * TILE_STRIDE + kbase] = v;    // conflict-free (stride 514)
  }
  __syncthreads();

  // WMMA: wave wv -> M-tile (wv&1), K-chunk (wv>>1) of 128
  const int lane  = t & 31;
  const int wv    = t >> 5;
  const int mtile = wv & 1;
  const int kc    = wv >> 1;
  const int lhalf = lane >> 4;             // upper lane half handles K+2,K+3
  const int lm    = lane & 15;
  const int arow  = mtile * 16 + lm;       // A: M = lane%16
  const int nrow  = (lm < 2) ? lm : 2;     // B: N = lane%16 (>=2 -> zero row)
  const float* aptr = &tile[arow * TILE_STRIDE + 2 * lhalf];
  const float* bptr = &wlds[nrow * 512 + 2 * lhalf];

  v8f cfr = {};
  const int k0 = kc * 128;
  #pragma unroll
  for (int k = 0; k < 128; k += 4) {
    v2f a  = *(const v2f*)(aptr + k0 + k);
    v2f bv = *(const v2f*)(bptr + k0 + k);
    cfr = __builtin_amdgcn_wmma_f32_16x16x4_f32(
        /*neg_a=*/false, a, /*neg_b=*/false, bv,
        /*c_mod=*/(short)0, cfr, /*reuse_a=*/false, /*reuse_b=*/false);
  }

  // C layout: lane 0/1 -> N=0/1, M=0..7 in v0..v7; lane 16/17 -> M=8..15
  if (lm < 2) {
    const int n  = lm;
    const int mb = mtile * 16 + lhalf * 8;
    #pragma unroll
    for (int j = 0; j < 8; ++j)
      atomicAdd(&acc[n * 32 + mb + j], cfr[j]);    // ds_add_f32 K-reduction
  }
  __syncthreads();

  if (t < 64) {
    const int n = t >> 5, wo = t & 31;
    y[(((size_t)(b * 2 + n) * 16 + d2) * 32 + h2) * 32 + wo] = acc[t];
  }
}

// ---------------------------------------------------------------------------
// Kernel 2: 2->1 channel 3x3x3 conv (pad 1) on [8,2,16,32,32], sigmoid,
// nearest 2x upsample = broadcast each coarse value to a 2x2x2 fine block.
// ---------------------------------------------------------------------------
__global__ __launch_bounds__(256) void conv_up_sig(
    const float* __restrict__ y, const float* __restrict__ cat_w,
    const float* __restrict__ cat_b, float* __restrict__ out)
{
  int idx = blockIdx.x * 256 + threadIdx.x;
  if (idx >= 8 * 16 * 32 * 32) return;
  const int w2 = idx & 31;
  const int h2 = (idx >> 5) & 31;
  const int d2 = (idx >> 10) & 15;
  const int b  = idx >> 14;

  float s = cat_b[0];
  #pragma unroll
  for (int n = 0; n < 2; ++n) {
    const float* yn = y + (size_t)(b * 2 + n) * (16 * 32 * 32);
    #pragma unroll
    for (int dd = -1; dd <= 1; ++dd) {
      int d = d2 + dd; if (d < 0 || d >= 16) continue;
      #pragma unroll
      for (int dh = -1; dh <= 1; ++dh) {
        int h = h2 + dh; if (h < 0 || h >= 32) continue;
        #pragma unroll
        for (int dw = -1; dw <= 1; ++dw) {
          int w = w2 + dw; if (w < 0 || w >= 32) continue;
          s += yn[(size_t)d * 1024 + h * 32 + w] *
               cat_w[((n * 3 + (dd + 1)) * 3 + (dh + 1)) * 3 + (dw + 1)];
        }
      }
    }
  }
  float sg = 1.0f / (1.0f + expf(-s));
  v2f v; v.x = sg; v.y = sg;
  float* ob = out + (size_t)b * (32 * 64 * 64)
                  + (size_t)(2 * d2) * 4096 + (2 * h2) * 64 + 2 * w2;
  *(v2f*)(ob)            = v;   // (kd=0,kh=0)
  *(v2f*)(ob + 64)       = v;   // (kd=0,kh=1)
  *(v2f*)(ob + 4096)     = v;   // (kd=1,kh=0)
  *(v2f*)(ob + 4096+64)  = v;   // (kd=1,kh=1)
}

extern "C" void kernel_launch(void* const* d_in, const int* in_sizes, int n_in,
                              void* d_out, int out_size, void* d_ws, size_t ws_size,
                              hipStream_t stream) {
  (void)in_sizes; (void)n_in; (void)out_size; (void)ws_size;
  const float* x      = (const float*)d_in[0];
  const float* mlp_w  = (const float*)d_in[1];
  const float* mlp_b  = (const float*)d_in[2];
  const float* conv_w = (const float*)d_in[3];
  const float* conv_b = (const float*)d_in[4];
  const float* cat_w  = (const float*)d_in[5];
  const float* cat_b  = (const float*)d_in[6];
  float* y   = (float*)d_ws;   // [8][2][16][32][32] f32 = 1 MiB scratch
  float* out = (float*)d_out;  // [8][1][32][64][64] f32

  patch_gemm_wmma<<<8 * 16 * 32, 256, 0, stream>>>(x, mlp_w, mlp_b, conv_w, conv_b, y);
  conv_up_sig<<<(8 * 16 * 32 * 32) / 256, 256, 0, stream>>>(y, cat_w, cat_b, out);
}